// cpss_50672024159018
// MI455X (gfx1250) — compile-verified
//
#include <hip/hip_runtime.h>
#include <hip/hip_bf16.h>
#include <math.h>

typedef __attribute__((ext_vector_type(2))) float v2f;
typedef __attribute__((ext_vector_type(8))) float v8f;

static __device__ __forceinline__ v8f wmma_f32(v2f a, v2f b, v8f c) {
  // D = A(16x4,f32) * B(4x16,f32) + C(16x16,f32)
  return __builtin_amdgcn_wmma_f32_16x16x4_f32(
      /*neg_a=*/false, a, /*neg_b=*/false, b,
      /*c_mod=*/(short)0, c, /*reuse_a=*/false, /*reuse_b=*/false);
}

// ---------------------------------------------------------------------------
// K1: conv3x3(fea_seg, w_seg) + bias -> maxpool2x2 -> tanh -> fea_aff
// fea_seg [32][64][160][160], w_seg [64][64][3][3], out fea_aff [32][64][80][80]
// Implicit GEMM: M = 16 pixels (4x4 spatial block), N = 16 channels, K = 576.
// Each workgroup: 8 waves = 2 M-tiles x 4 N-tiles; weights staged in LDS.
// Pool fusion: D-matrix rows M={0,1,4,5}->pooled px0, {2,3,6,7}->px1, per half.
// ---------------------------------------------------------------------------
__global__ __launch_bounds__(256) void k_conv_seg(
    const float* __restrict__ x, const float* __restrict__ w,
    const float* __restrict__ bias, float* __restrict__ fea_aff) {
  __shared__ float wlds[576 * 64];  // [k][n], 144 KB
  const int tid = threadIdx.x;
  for (int idx = tid; idx < 576 * 64; idx += 256) {
    int k = idx >> 6, n = idx & 63;
    int cin = k / 9, r = k % 9;
    wlds[idx] = w[(n * 64 + cin) * 9 + r];
  }
  __syncthreads();

  const int wave = tid >> 5, lane = tid & 31;
  const int mt = wave & 1, nt = wave >> 1;
  const long gtile = (long)blockIdx.x * 2 + mt;  // 0..51199
  const int b = (int)(gtile / 1600);
  const int t = (int)(gtile % 1600);
  const int ty = t / 40, tx = t % 40;
  const int y0 = ty * 4, x0 = tx * 4;

  const int m = lane & 15;           // A-row / B-col index
  const int hi = lane >> 4;
  const int khalf = hi * 2;          // lanes 0-15: K0,K1 ; 16-31: K2,K3
  const int py = y0 + (m >> 2), px = x0 + (m & 3);
  const float* xb = x + (long)b * 64 * 160 * 160;

  v8f acc = {};
  for (int ks = 0; ks < 576; ks += 4) {
    v2f a = {}, bb = {};
#pragma unroll
    for (int u = 0; u < 2; ++u) {
      int k = ks + khalf + u;
      int cin = k / 9, r = k % 9;
      int yy = py + (r / 3) - 1, xx = px + (r % 3) - 1;
      float av = 0.f;
      if (yy >= 0 && yy < 160 && xx >= 0 && xx < 160)
        av = xb[(long)cin * 25600 + yy * 160 + xx];
      a[u] = av;
      bb[u] = wlds[k * 64 + nt * 16 + m];
    }
    acc = wmma_f32(a, bb, acc);
  }

  // bias + 2x2 maxpool + tanh (max/tanh monotone: bias after max is exact)
  const int ch = nt * 16 + m;
  const float cb = bias[ch];
  float p0 = fmaxf(fmaxf(acc[0], acc[1]), fmaxf(acc[4], acc[5]));
  float p1 = fmaxf(fmaxf(acc[2], acc[3]), fmaxf(acc[6], acc[7]));
  p0 = tanhf(p0 + cb);
  p1 = tanhf(p1 + cb);
  const int pooly = ty * 2 + hi, poolx = tx * 2;
  float* o = fea_aff + (((long)b * 64 + ch) * 80 + pooly) * 80 + poolx;
  o[0] = p0;
  o[1] = p1;
}

// ---------------------------------------------------------------------------
// K2: conv3x3(fea_class, w_cls), no bias -> fc_raw [32][64][40][40]
// K = 2304 processed in 4 LDS-staged chunks of 576 (Cin 64 at a time).
// M tile = 2 rows x 8 cols of the 40x40 plane.
// ---------------------------------------------------------------------------
__global__ __launch_bounds__(256) void k_conv_cls(
    const float* __restrict__ x, const float* __restrict__ w,
    float* __restrict__ out) {
  __shared__ float wlds[576 * 64];  // 144 KB, reloaded per Cin-chunk
  const int tid = threadIdx.x, wave = tid >> 5, lane = tid & 31;
  const int mt = wave & 1, nt = wave >> 1;
  const long gtile = (long)blockIdx.x * 2 + mt;  // 0..3199
  const int b = (int)(gtile / 100);
  const int t = (int)(gtile % 100);
  const int ty = t / 5, tx = t % 5;
  const int y0 = ty * 2, x0 = tx * 8;

  const int m = lane & 15;
  const int hi = lane >> 4;
  const int khalf = hi * 2;
  const int py = y0 + (m >> 3), px = x0 + (m & 7);
  const float* xb = x + (long)b * 256 * 1600;

  v8f acc = {};
  for (int cc = 0; cc < 256; cc += 64) {
    __syncthreads();
    for (int idx = tid; idx < 576 * 64; idx += 256) {
      int k = idx >> 6, n = idx & 63;
      int cin = cc + k / 9, r = k % 9;
      wlds[idx] = w[((long)n * 256 + cin) * 9 + r];
    }
    __syncthreads();
    for (int ks = 0; ks < 576; ks += 4) {
      v2f a = {}, bb = {};
#pragma unroll
      for (int u = 0; u < 2; ++u) {
        int k = ks + khalf + u;
        int cin = cc + k / 9, r = k % 9;
        int yy = py + (r / 3) - 1, xx = px + (r % 3) - 1;
        float av = 0.f;
        if (yy >= 0 && yy < 40 && xx >= 0 && xx < 40)
          av = xb[(long)cin * 1600 + yy * 40 + xx];
        a[u] = av;
        bb[u] = wlds[k * 64 + nt * 16 + m];
      }
      acc = wmma_f32(a, bb, acc);
    }
  }
  const int ch = nt * 16 + m;
#pragma unroll
  for (int v = 0; v < 8; ++v) {
    int M = v + 8 * hi;
    int yy = y0 + (M >> 3), xx = x0 + (M & 7);
    out[(((long)b * 64 + ch) * 40 + yy) * 40 + xx] = acc[v];
  }
}

// ---------------------------------------------------------------------------
// K3: per-channel train-mode BN statistics. x: [N][C][plane]. stats[C][2]=(mean, rstd)
// One workgroup per channel.
// ---------------------------------------------------------------------------
__global__ __launch_bounds__(256) void k_bnstats(
    const float* __restrict__ x, float* __restrict__ stats,
    int C, int plane, int N) {
  const int ch = blockIdx.x;
  float s = 0.f, s2 = 0.f;
  const long total = (long)N * plane;
  for (long i = threadIdx.x; i < total; i += blockDim.x) {
    int b = (int)(i / plane);
    int off = (int)(i % plane);
    float v = x[((long)b * C + ch) * plane + off];
    s += v;
    s2 += v * v;
  }
  __shared__ float sh0[256], sh1[256];
  sh0[threadIdx.x] = s;
  sh1[threadIdx.x] = s2;
  __syncthreads();
  for (int st = 128; st > 0; st >>= 1) {
    if ((int)threadIdx.x < st) {
      sh0[threadIdx.x] += sh0[threadIdx.x + st];
      sh1[threadIdx.x] += sh1[threadIdx.x + st];
    }
    __syncthreads();
  }
  if (threadIdx.x == 0) {
    float cnt = (float)total;
    float mean = sh0[0] / cnt;
    float var = sh1[0] / cnt - mean * mean;
    stats[ch * 2 + 0] = mean;
    stats[ch * 2 + 1] = rsqrtf(var + 1e-5f);
  }
}

// ---------------------------------------------------------------------------
// K4 (mega, one workgroup per (b,i,j) block, 800 WGs, 152 KB LDS):
//   phase A: affinity = A^T A  (A = 64x256 tanh block), written to d_out,
//            2x2 q-pooled aff_dn accumulated in registers -> LDS (no atomics)
//   phase B: masked = aff_dn(256x64) x fcb(64x64)  (fcb = BN+ReLU of fc, staged)
//   phase C: seg_raw = masked(256x64) x w_head^T(64x21) + b_head -> d_out
// ---------------------------------------------------------------------------
__global__ __launch_bounds__(256) void k_block(
    const float* __restrict__ fea_aff,   // [32][64][80][80]
    const float* __restrict__ fc_raw,    // [32][64][40][40]
    const float* __restrict__ st_cls,    // [64][2]
    const float* __restrict__ g_cls, const float* __restrict__ beta_cls,
    const float* __restrict__ w_head,    // [21][64]
    const float* __restrict__ b_head,    // [21]
    float* __restrict__ aff_out,         // [32][5][5][256][256]
    float* __restrict__ seg_raw) {       // [32][21][80][80]
  __shared__ float sA[64 * 256];    // [c][p]  64 KB ; reused as sM[p][c] later
  __shared__ float sPool[256 * 64]; // aff_dn [p][qd] 64 KB
  __shared__ float sFcb[64 * 64];   // [qd][c] 16 KB
  __shared__ float sWh[64 * 32];    // [c][nn] 8 KB (nn>=21 zero-padded)
  float* sM = sA;                   // masked [p][c], reuses sA after barrier

  const int tid = threadIdx.x, wave = tid >> 5, lane = tid & 31;
  const int n = lane & 15, hi = lane >> 4, khalf = hi * 2;
  const int b = blockIdx.x / 25;
  const int ij = blockIdx.x % 25;
  const int i = ij / 5, j = ij % 5;

  // stage A block (tanh features), fcb (BN+ReLU applied), head weights
  for (int idx = tid; idx < 64 * 256; idx += 256) {
    int c = idx >> 8, p = idx & 255;
    sA[idx] = fea_aff[(((long)b * 64 + c) * 80 + i * 16 + (p >> 4)) * 80 +
                      j * 16 + (p & 15)];
  }
  for (int idx = tid; idx < 64 * 64; idx += 256) {
    int qd = idx >> 6, c = idx & 63;
    float raw = fc_raw[(((long)b * 64 + c) * 40 + i * 8 + (qd >> 3)) * 40 +
                       j * 8 + (qd & 7)];
    float v = (raw - st_cls[c * 2]) * st_cls[c * 2 + 1] * g_cls[c] + beta_cls[c];
    sFcb[idx] = fmaxf(v, 0.f);
  }
  for (int idx = tid; idx < 64 * 32; idx += 256) {
    int c = idx >> 5, nn = idx & 31;
    sWh[idx] = (nn < 21) ? w_head[nn * 64 + c] : 0.f;
  }
  __syncthreads();

  // ---- phase A: affinity (256 tiles as 128 qt-pairs) ----
  float* ob = aff_out + ((long)blockIdx.x << 16);  // 256*256 per block
  for (int ti = wave; ti < 128; ti += 8) {
    const int pt = ti >> 3;   // 0..15
    const int qtp = ti & 7;   // qt pair 0..7
    float pooled[8];
#pragma unroll
    for (int v = 0; v < 8; ++v) pooled[v] = 0.f;
#pragma unroll
    for (int half = 0; half < 2; ++half) {
      const int qt = qtp * 2 + half;
      v8f acc = {};
      for (int ks = 0; ks < 64; ks += 4) {
        v2f a = {}, bb = {};
#pragma unroll
        for (int u = 0; u < 2; ++u) {
          int k = ks + khalf + u;
          a[u] = sA[k * 256 + pt * 16 + n];
          bb[u] = sA[k * 256 + qt * 16 + n];
        }
        acc = wmma_f32(a, bb, acc);
      }
#pragma unroll
      for (int v = 0; v < 8; ++v) {
        int p = pt * 16 + v + 8 * hi;
        ob[p * 256 + qt * 16 + n] = acc[v];
        pooled[v] += acc[v];
      }
    }
    // fold the qx neighbor (lanes n, n^1) -> aff_dn entry, one writer per slot
#pragma unroll
    for (int v = 0; v < 8; ++v) {
      float s = pooled[v] + __shfl_xor(pooled[v], 1);
      if ((n & 1) == 0) {
        int p = pt * 16 + v + 8 * hi;
        sPool[p * 64 + qtp * 8 + (n >> 1)] = 0.25f * s;
      }
    }
  }
  __syncthreads();

  // ---- phase B: masked[p][c] = aff_dn x fcb  (M=256,N=64,K=64) ----
  for (int ti = wave; ti < 64; ti += 8) {
    const int pt = ti >> 2, ct = ti & 3;
    v8f acc = {};
    for (int ks = 0; ks < 64; ks += 4) {
      v2f a = {}, bb = {};
#pragma unroll
      for (int u = 0; u < 2; ++u) {
        int k = ks + khalf + u;
        a[u] = sPool[(pt * 16 + n) * 64 + k];
        bb[u] = sFcb[k * 64 + ct * 16 + n];
      }
      acc = wmma_f32(a, bb, acc);
    }
#pragma unroll
    for (int v = 0; v < 8; ++v)
      sM[(pt * 16 + v + 8 * hi) * 64 + ct * 16 + n] = acc[v];
  }
  __syncthreads();

  // ---- phase C: head 1x1 conv (M=256, N=21 in 2 tiles, K=64) ----
  for (int ti = wave; ti < 32; ti += 8) {
    const int pt = ti >> 1, ntile = ti & 1;
    v8f acc = {};
    for (int ks = 0; ks < 64; ks += 4) {
      v2f a = {}, bb = {};
#pragma unroll
      for (int u = 0; u < 2; ++u) {
        int k = ks + khalf + u;
        a[u] = sM[(pt * 16 + n) * 64 + k];
        bb[u] = sWh[k * 32 + ntile * 16 + n];
      }
      acc = wmma_f32(a, bb, acc);
    }
    const int nc = ntile * 16 + n;
#pragma unroll
    for (int v = 0; v < 8; ++v) {
      if (nc < 21) {
        int p = pt * 16 + v + 8 * hi;
        seg_raw[(((long)b * 21 + nc) * 80 + i * 16 + (p >> 4)) * 80 + j * 16 +
                (p & 15)] = acc[v] + b_head[nc];
      }
    }
  }
}

// ---------------------------------------------------------------------------
// K5: apply BN in place on seg. x: [N][C][plane]
// ---------------------------------------------------------------------------
__global__ __launch_bounds__(256) void k_bnapply(
    float* __restrict__ x, const float* __restrict__ stats,
    const float* __restrict__ g, const float* __restrict__ beta,
    int C, int plane, long total) {
  long idx = (long)blockIdx.x * 256 + threadIdx.x;
  if (idx >= total) return;
  int ch = (int)((idx / plane) % C);
  x[idx] = (x[idx] - stats[ch * 2]) * stats[ch * 2 + 1] * g[ch] + beta[ch];
}

// ---------------------------------------------------------------------------
extern "C" void kernel_launch(void* const* d_in, const int* in_sizes, int n_in,
                              void* d_out, int out_size, void* d_ws,
                              size_t ws_size, hipStream_t stream) {
  const float* fea_seg   = (const float*)d_in[0];
  const float* fea_class = (const float*)d_in[1];
  const float* w_seg     = (const float*)d_in[2];
  const float* b_seg     = (const float*)d_in[3];
  const float* w_cls     = (const float*)d_in[4];
  const float* g_cls     = (const float*)d_in[5];
  const float* beta_cls  = (const float*)d_in[6];
  const float* w_head    = (const float*)d_in[7];
  const float* b_head    = (const float*)d_in[8];
  const float* g_head    = (const float*)d_in[9];
  const float* beta_head = (const float*)d_in[10];

  float* seg = (float*)d_out;                 // [32][21][80][80] = 4,300,800
  float* aff = seg + 4300800;                 // [32][5][5][256][256]

  float* ws      = (float*)d_ws;
  float* fea_aff = ws;                        // 13,107,200 f
  float* fc_raw  = ws + 13107200;             //  3,276,800 f
  float* st_cls  = ws + 16384000;             //        128 f
  float* st_head = ws + 16384128;             //         64 f

  k_conv_seg<<<25600, 256, 0, stream>>>(fea_seg, w_seg, b_seg, fea_aff);
  k_conv_cls<<<1600, 256, 0, stream>>>(fea_class, w_cls, fc_raw);
  k_bnstats<<<64, 256, 0, stream>>>(fc_raw, st_cls, 64, 1600, 32);
  k_block<<<800, 256, 0, stream>>>(fea_aff, fc_raw, st_cls, g_cls, beta_cls,
                                   w_head, b_head, aff, seg);
  k_bnstats<<<21, 256, 0, stream>>>(seg, st_head, 21, 6400, 32);
  const long seg_total = 4300800;
  k_bnapply<<<(int)((seg_total + 255) / 256), 256, 0, stream>>>(
      seg, st_head, g_head, beta_head, 21, 6400, seg_total);
}